// SetCriterion_56246891708455
// MI455X (gfx1250) — compile-verified
//
#include <hip/hip_runtime.h>
#include <math.h>

// ---------------- constants ----------------
#define NCLS   80
#define BATCH  16
#define NANCH  3
#define CH     85           // 5 + NCLS

// Workspace layout (in floats)
#define TOBJ0   0            // 16*3*80*80 = 307200
#define TOBJ1   307200       // 16*3*40*40 = 76800
#define TOBJ2   384000       // 16*3*20*20 = 19200
#define LBOX_P  403200       // 32 slots (16 + 8 + 4 used)
#define LCLS_P  403232       // 32 slots
#define LOBJ_P  403264       // 1600 slots (1200 + 300 + 75 used)
#define WS_TOTAL 404864

typedef __attribute__((ext_vector_type(2))) float v2f;
typedef __attribute__((ext_vector_type(8))) float v8f;

// ---------------- wave reduction via f32 WMMA ----------------
// D = A x B + 0 with B = ones(4x16): D[m][n] = sum_k A[m][k].
// Lane L contributes its value v exactly once into row (L % 16) of A
// (a.y = 0 zeroes the other K slot). Row sums are replicated across N, so
// lanes 0-15 see rowsums 0..7 in d[0..7], lanes 16-31 see rowsums 8..15.
// Summing d[0..7] in-lane and folding across half-waves yields the full
// 32-lane sum in every lane. Exact f32, requires EXEC == all ones.
__device__ __forceinline__ float wave_reduce_wmma(float v) {
    v2f a; a.x = v;    a.y = 0.0f;
    v2f b; b.x = 1.0f; b.y = 1.0f;
    v8f c = {};
    c = __builtin_amdgcn_wmma_f32_16x16x4_f32(
        /*neg_a=*/false, a, /*neg_b=*/false, b,
        /*c_mod=*/(short)0, c, /*reuse_a=*/false, /*reuse_b=*/false);
    float s = ((c[0] + c[1]) + (c[2] + c[3])) + ((c[4] + c[5]) + (c[6] + c[7]));
    s += __shfl_xor(s, 16, 32);
    return s;
}

__device__ __forceinline__ float softplusf(float x) {
    // logaddexp(0, x) = max(x,0) + log1p(exp(-|x|)), numerically stable
    return fmaxf(x, 0.0f) + log1pf(expf(-fabsf(x)));
}

__device__ __forceinline__ float sigmoidf(float x) {
    return 1.0f / (1.0f + expf(-x));
}

// ---------------- kernels ----------------
__global__ void zero_kernel(float* __restrict__ ws, int n) {
    int i = blockIdx.x * blockDim.x + threadIdx.x;
    if (i < n) ws[i] = 0.0f;
}

// One thread per target. Grid is exactly n/256 blocks (n is a multiple of 256),
// so EXEC is all-ones everywhere (WMMA-safe).
__global__ void gather_kernel(const float* __restrict__ p,
                              const float* __restrict__ tbox,
                              const float* __restrict__ anch,
                              const int* __restrict__ bb,
                              const int* __restrict__ aa,
                              const int* __restrict__ gj,
                              const int* __restrict__ gi,
                              const int* __restrict__ tcls,
                              float* __restrict__ tobj,
                              float* __restrict__ lbox_part,
                              float* __restrict__ lcls_part,
                              int g) {
    const int i = blockIdx.x * blockDim.x + threadIdx.x;

    const int b  = bb[i];
    const int a  = aa[i];
    const int j  = gj[i];
    const int k  = gi[i];
    const int tc = tcls[i];

    const int cell = ((b * NANCH + a) * g + j) * g + k;
    const float* __restrict__ ps = p + (size_t)cell * CH;

    // decode predicted box
    const float px = sigmoidf(ps[0]) * 2.0f - 0.5f;
    const float py = sigmoidf(ps[1]) * 2.0f - 0.5f;
    float sw = sigmoidf(ps[2]) * 2.0f;
    float sh = sigmoidf(ps[3]) * 2.0f;
    const float pw = sw * sw * anch[2 * i + 0];
    const float ph = sh * sh * anch[2 * i + 1];

    const float x2 = tbox[4 * i + 0];
    const float y2 = tbox[4 * i + 1];
    const float w2 = tbox[4 * i + 2];
    const float h2 = tbox[4 * i + 3];

    // CIoU (matches reference element-for-element)
    const float eps = 1e-7f;
    const float b1x1 = px - pw * 0.5f, b1x2 = px + pw * 0.5f;
    const float b1y1 = py - ph * 0.5f, b1y2 = py + ph * 0.5f;
    const float b2x1 = x2 - w2 * 0.5f, b2x2 = x2 + w2 * 0.5f;
    const float b2y1 = y2 - h2 * 0.5f, b2y2 = y2 + h2 * 0.5f;

    float iw = fminf(b1x2, b2x2) - fmaxf(b1x1, b2x1); iw = fmaxf(iw, 0.0f);
    float ih = fminf(b1y2, b2y2) - fmaxf(b1y1, b2y1); ih = fmaxf(ih, 0.0f);
    const float inter = iw * ih;
    const float uni   = pw * ph + w2 * h2 - inter + eps;
    const float iou   = inter / uni;

    const float cw = fmaxf(b1x2, b2x2) - fminf(b1x1, b2x1);
    const float chh = fmaxf(b1y2, b2y2) - fminf(b1y1, b2y1);
    const float c2 = cw * cw + chh * chh + eps;
    const float dx = b2x1 + b2x2 - b1x1 - b1x2;
    const float dy = b2y1 + b2y2 - b1y1 - b1y2;
    const float rho2 = (dx * dx + dy * dy) * 0.25f;
    const float dat = atanf(w2 / (h2 + eps)) - atanf(pw / (ph + eps));
    const float v = 0.40528473456935109f * dat * dat;   // 4/pi^2
    const float alpha = v / (v - iou + (1.0f + eps));
    const float ciou = iou - (rho2 / c2 + v * alpha);

    const float lbox_term = 1.0f - ciou;

    // scatter tobj = clip(ciou, 0); atomicMax on nonneg float bits is
    // order-preserving and deterministic under duplicate indices.
    const float tv = fmaxf(ciou, 0.0f);
    atomicMax((unsigned int*)(tobj + cell), __float_as_uint(tv));

    // class BCE: sum_c softplus(x_c) - x_{tcls}
    float lcls_acc = 0.0f;
    #pragma unroll 8
    for (int c = 0; c < NCLS; ++c) {
        lcls_acc += softplusf(ps[5 + c]);
    }
    lcls_acc -= ps[5 + tc];

    // block reduction: WMMA per-wave, LDS across the 8 waves
    __shared__ float red_b[8];
    __shared__ float red_c[8];
    const float wb = wave_reduce_wmma(lbox_term);
    const float wc = wave_reduce_wmma(lcls_acc);
    const int lane = threadIdx.x & 31;
    const int wv   = threadIdx.x >> 5;
    if (lane == 0) { red_b[wv] = wb; red_c[wv] = wc; }
    __syncthreads();
    if (threadIdx.x == 0) {
        float sb = 0.0f, sc = 0.0f;
        #pragma unroll
        for (int w = 0; w < 8; ++w) { sb += red_b[w]; sc += red_c[w]; }
        lbox_part[blockIdx.x] = sb;
        lcls_part[blockIdx.x] = sc;
    }
}

// One thread per objectness element. M is a multiple of 256 -> full EXEC.
__global__ void obj_kernel(const float* __restrict__ p,
                           const float* __restrict__ tobj,
                           float* __restrict__ lobj_part) {
    const int e = blockIdx.x * blockDim.x + threadIdx.x;
    const float x = p[(size_t)e * CH + 4];
    const float t = tobj[e];
    const float v = softplusf(x) - x * t;

    __shared__ float red[8];
    const float wsum = wave_reduce_wmma(v);
    const int lane = threadIdx.x & 31;
    const int wv   = threadIdx.x >> 5;
    if (lane == 0) red[wv] = wsum;
    __syncthreads();
    if (threadIdx.x == 0) {
        float s = 0.0f;
        #pragma unroll
        for (int w = 0; w < 8; ++w) s += red[w];
        lobj_part[blockIdx.x] = s;
    }
}

// Single thread finalize: serial (deterministic) sum of ~1.6K partials.
__global__ void final_kernel(const float* __restrict__ ws, float* __restrict__ out) {
    const int   nb_box[3]  = {16, 8, 4};
    const int   off_box[3] = {0, 16, 24};
    const int   nb_obj[3]  = {1200, 300, 75};
    const int   off_obj[3] = {0, 1200, 1500};
    const float n_t[3]     = {4096.0f, 2048.0f, 1024.0f};
    const float M_t[3]     = {307200.0f, 76800.0f, 19200.0f};
    const float bal[3]     = {4.0f, 1.0f, 0.4f};

    float lbox = 0.0f, lcls = 0.0f, lobj = 0.0f;
    for (int l = 0; l < 3; ++l) {
        float sb = 0.0f, sc = 0.0f, so = 0.0f;
        for (int i = 0; i < nb_box[l]; ++i) {
            sb += ws[LBOX_P + off_box[l] + i];
            sc += ws[LCLS_P + off_box[l] + i];
        }
        for (int i = 0; i < nb_obj[l]; ++i) so += ws[LOBJ_P + off_obj[l] + i];
        lbox += sb / n_t[l];
        lcls += sc / (n_t[l] * (float)NCLS);
        lobj += (so / M_t[l]) * bal[l];
    }
    lbox *= 0.05f;   // W_BOX
    lobj *= 1.0f;    // W_OBJ
    lcls *= 0.5f;    // W_CLS
    const float loss = lbox + lobj + lcls;
    out[0] = loss * (float)BATCH;
    out[1] = lbox;
    out[2] = lobj;
    out[3] = lcls;
    out[4] = loss;
}

// ---------------- launch ----------------
extern "C" void kernel_launch(void* const* d_in, const int* in_sizes, int n_in,
                              void* d_out, int out_size, void* d_ws, size_t ws_size,
                              hipStream_t stream) {
    (void)in_sizes; (void)n_in; (void)out_size; (void)ws_size;
    float* ws = (float*)d_ws;

    zero_kernel<<<(WS_TOTAL + 255) / 256, 256, 0, stream>>>(ws, WS_TOTAL);

    const int gs[3]       = {80, 40, 20};
    const int ns[3]       = {4096, 2048, 1024};
    const int tobj_off[3] = {TOBJ0, TOBJ1, TOBJ2};
    const int box_off[3]  = {0, 16, 24};
    const int obj_off[3]  = {0, 1200, 1500};

    for (int l = 0; l < 3; ++l) {
        const float* p  = (const float*)d_in[l * 8 + 0];
        const float* tb = (const float*)d_in[l * 8 + 1];
        const float* an = (const float*)d_in[l * 8 + 2];
        const int*   bb = (const int*)d_in[l * 8 + 3];
        const int*   aa = (const int*)d_in[l * 8 + 4];
        const int*   gj = (const int*)d_in[l * 8 + 5];
        const int*   gi = (const int*)d_in[l * 8 + 6];
        const int*   tc = (const int*)d_in[l * 8 + 7];
        gather_kernel<<<ns[l] / 256, 256, 0, stream>>>(
            p, tb, an, bb, aa, gj, gi, tc,
            ws + tobj_off[l],
            ws + LBOX_P + box_off[l],
            ws + LCLS_P + box_off[l],
            gs[l]);
    }

    for (int l = 0; l < 3; ++l) {
        const float* p = (const float*)d_in[l * 8 + 0];
        const int M = BATCH * NANCH * gs[l] * gs[l];
        obj_kernel<<<M / 256, 256, 0, stream>>>(p, ws + tobj_off[l],
                                                ws + LOBJ_P + obj_off[l]);
    }

    final_kernel<<<1, 1, 0, stream>>>(ws, (float*)d_out);
}